// LSEPLoss_13572096655779
// MI455X (gfx1250) — compile-verified
//
#include <hip/hip_runtime.h>
#include <math.h>

typedef __attribute__((ext_vector_type(2))) float v2f;
typedef __attribute__((ext_vector_type(8))) float v8f;
typedef __attribute__((ext_vector_type(2))) int   v2i;

#define NROWS 16384
#define NCOLS 2048

// Block: 256 threads = 8 wave32s. Each block owns 16 rows of the matrix.
// Wave w reduces columns [w*256, (w+1)*256) of those 16 rows using
// V_WMMA_F32_16X16X4_F32 with an all-ones B matrix as a row-sum engine:
//   D[m,0] += sum_k A[m,k] * 1
__global__ __launch_bounds__(256)
void lsep_row_kernel(const float* __restrict__ pred,
                     const int*   __restrict__ lab,
                     float* __restrict__ row_loss) {
    const int tid   = threadIdx.x;
    const int lane  = tid & 31;
    const int wave  = tid >> 5;            // 0..7
    const int m     = lane & 15;           // row within 16-row tile
    const int khalf = (lane >> 4) << 1;    // 0 for lanes 0-15, 2 for lanes 16-31

    const long row_base = (long)blockIdx.x * 16;
    const long row      = row_base + m;
    const float* prow = pred + row * (long)NCOLS;
    const int*   lrow = lab  + row * (long)NCOLS;

    v8f c_neg = {0.f,0.f,0.f,0.f,0.f,0.f,0.f,0.f};
    v8f c_pos = {0.f,0.f,0.f,0.f,0.f,0.f,0.f,0.f};
    const v2f b_ones = {1.0f, 1.0f};

    const int col0 = wave * 256;

    #pragma unroll 4
    for (int k0 = 0; k0 < 256; k0 += 4) {
        const int c = col0 + k0 + khalf;       // even -> 8B aligned
        v2f xv = *(const v2f*)(prow + c);      // A-layout: this lane's K, K+1
        v2i lv = *(const v2i*)(lrow + c);

        float e0p = __expf(xv.x), e0n = __expf(-xv.x);
        float e1p = __expf(xv.y), e1n = __expf(-xv.y);

        v2f a_neg, a_pos;
        a_neg.x = lv.x ? 0.0f : e0p;   // negatives contribute exp(x)
        a_neg.y = lv.y ? 0.0f : e1p;
        a_pos.x = lv.x ? e0n : 0.0f;   // positives contribute exp(-x)
        a_pos.y = lv.y ? e1n : 0.0f;

        // D = A x ones + C : accumulates per-row sums over this K slice.
        c_neg = __builtin_amdgcn_wmma_f32_16x16x4_f32(
            false, a_neg, false, b_ones, (short)0, c_neg, false, false);
        c_pos = __builtin_amdgcn_wmma_f32_16x16x4_f32(
            false, a_pos, false, b_ones, (short)0, c_pos, false, false);
    }

    // D[m, n=0] lives in lane 0 (rows 0..7 across VGPR 0..7) and
    // lane 16 (rows 8..15 across VGPR 0..7).
    __shared__ float s_neg[8][16];
    __shared__ float s_pos[8][16];
    if (lane == 0 || lane == 16) {
        const int mbase = (lane == 0) ? 0 : 8;
        #pragma unroll
        for (int i = 0; i < 8; ++i) {
            s_neg[wave][mbase + i] = c_neg[i];
            s_pos[wave][mbase + i] = c_pos[i];
        }
    }
    __syncthreads();

    // Combine the 8 waves' column-range partials; one loss per row.
    if (wave == 0 && lane < 16) {
        float ns = 0.0f, ps = 0.0f;
        #pragma unroll
        for (int w = 0; w < 8; ++w) {
            ns += s_neg[w][lane];
            ps += s_pos[w][lane];
        }
        row_loss[row_base + lane] = log1pf(ns * ps);
    }
}

// Deterministic single-block mean of the 16384 per-row losses.
__global__ __launch_bounds__(256)
void lsep_mean_kernel(const float* __restrict__ row_loss,
                      float* __restrict__ out) {
    __shared__ float s[256];
    float acc = 0.0f;
    for (int i = threadIdx.x; i < NROWS; i += 256)
        acc += row_loss[i];
    s[threadIdx.x] = acc;
    __syncthreads();
    #pragma unroll
    for (int off = 128; off > 0; off >>= 1) {
        if (threadIdx.x < off) s[threadIdx.x] += s[threadIdx.x + off];
        __syncthreads();
    }
    if (threadIdx.x == 0)
        out[0] = s[0] * (1.0f / (float)NROWS);
}

extern "C" void kernel_launch(void* const* d_in, const int* in_sizes, int n_in,
                              void* d_out, int out_size, void* d_ws, size_t ws_size,
                              hipStream_t stream) {
    const float* pred = (const float*)d_in[0];
    const int*   lab  = (const int*)d_in[1];
    float* row_loss   = (float*)d_ws;        // 16384 floats of scratch

    lsep_row_kernel<<<NROWS / 16, 256, 0, stream>>>(pred, lab, row_loss);
    lsep_mean_kernel<<<1, 256, 0, stream>>>(row_loss, (float*)d_out);
}